// GWNN_41970420418155
// MI455X (gfx1250) — compile-verified
//
#include <hip/hip_runtime.h>
#include <math.h>

// ---------------------------------------------------------------------------
// GWNN forward on MI455X (gfx1250, wave32).
//   h = in_feat @ W1 + b1                      -> fp32 WMMA 16x16x4
//   h = gconv(h, g)   (both-norm, edge weight) -> L2-resident scatter-add
//   h = kernel * h                             -> fused with both norms
//   h = gconv(h, ginv)                         -> L2-resident scatter-add
//   out = relu(h) @ W3 + b3                    -> wave32 shuffle reduction
// ---------------------------------------------------------------------------

typedef __attribute__((ext_vector_type(2))) float v2f;
typedef __attribute__((ext_vector_type(8))) float v8f;

#define IN_F 128
#define H_F  64
#define C_F  2

// ---- degree accumulation: 4 atomic bumps per edge -------------------------
__global__ __launch_bounds__(256) void deg_kernel(
    const int* __restrict__ gs, const int* __restrict__ gd,
    const int* __restrict__ is_, const int* __restrict__ id_,
    float* __restrict__ deg, int Nn, int Ee)
{
    int t = blockIdx.x * 256 + threadIdx.x;
    if (t >= Ee) return;
    int a = __builtin_nontemporal_load(gs + t);
    int b = __builtin_nontemporal_load(gd + t);
    int c = __builtin_nontemporal_load(is_ + t);
    int d = __builtin_nontemporal_load(id_ + t);
    atomicAdd(deg + a,            1.0f);   // deg_out(g)
    atomicAdd(deg + Nn + b,       1.0f);   // deg_in(g)
    atomicAdd(deg + 2 * Nn + c,   1.0f);   // deg_out(ginv)
    atomicAdd(deg + 3 * Nn + d,   1.0f);   // deg_in(ginv)
}

// ---- deg -> rsqrt(max(deg,1)) in place ------------------------------------
__global__ __launch_bounds__(256) void deg_finish_kernel(float* __restrict__ deg, int n4)
{
    int t = blockIdx.x * 256 + threadIdx.x;
    if (t >= n4) return;
    deg[t] = rsqrtf(fmaxf(deg[t], 1.0f));
}

// ---- GEMM1: Hout[row,:] = (X @ W1 + b1)[row,:] * rs_dout[row] --------------
// One wave per 16x64 tile.  fp32 WMMA 16x16x4, K stepped by 4 (32 steps).
// A frag (16x4 f32, ISA 7.12.2): lane&15 = M row; lane>>4 selects K pair
// {0,1} vs {2,3}; v2f holds the K pair.  B frag mirrors with lane&15 = N col.
// D layout: VGPR v holds M = v + 8*(lane>>4), N = lane&15.
__global__ __launch_bounds__(256) void gemm1_wmma_kernel(
    const float* __restrict__ X,        // [N,128]
    const float* __restrict__ W1,       // [128,64]
    const float* __restrict__ b1,       // [64]
    const float* __restrict__ rs_dout,  // [N] = deg_out(g)^{-1/2}
    float* __restrict__ Hout,           // [N,64]
    int Nn)
{
    const int lane  = threadIdx.x & 31;
    const int wave  = threadIdx.x >> 5;
    const int tile  = blockIdx.x * 8 + wave;        // 16-row tile
    const int ntile = (Nn + 15) >> 4;
    if (tile >= ntile) return;                      // wave-uniform: EXEC stays full

    const int n0    = lane & 15;
    const int half  = lane >> 4;
    const int m     = (tile << 4) + n0;             // A row for this lane
    const int mclmp = m < Nn ? m : (Nn - 1);
    const int kh    = half << 1;                    // K offset within step: 0 or 2

    const float* xrow = X + (size_t)mclmp * IN_F;

    v8f acc0 = {}, acc1 = {}, acc2 = {}, acc3 = {};

    for (int k = 0; k < IN_F; k += 4) {
        const int ka = k + kh;
        v2f a;  a.x = xrow[ka];  a.y = xrow[ka + 1];
        const float* wr0 = W1 + (size_t)ka * H_F;       // row ka   of W1
        const float* wr1 = wr0 + H_F;                   // row ka+1 of W1
        v2f b0; b0.x = wr0[n0];      b0.y = wr1[n0];
        v2f b1v;b1v.x = wr0[16+n0];  b1v.y = wr1[16+n0];
        v2f b2; b2.x = wr0[32+n0];   b2.y = wr1[32+n0];
        v2f b3v;b3v.x = wr0[48+n0];  b3v.y = wr1[48+n0];
        acc0 = __builtin_amdgcn_wmma_f32_16x16x4_f32(false, a, false, b0,  (short)0, acc0, false, false);
        acc1 = __builtin_amdgcn_wmma_f32_16x16x4_f32(false, a, false, b1v, (short)0, acc1, false, false);
        acc2 = __builtin_amdgcn_wmma_f32_16x16x4_f32(false, a, false, b2,  (short)0, acc2, false, false);
        acc3 = __builtin_amdgcn_wmma_f32_16x16x4_f32(false, a, false, b3v, (short)0, acc3, false, false);
    }

    // Epilogue: bias + D_out^{-1/2} prescale fused into the store.
#pragma unroll
    for (int v = 0; v < 8; ++v) {
        const int row = (tile << 4) + v + half * 8;
        if (row < Nn) {
            const float s = rs_dout[row];
            float* hr = Hout + (size_t)row * H_F;
            hr[ 0 + n0] = (acc0[v] + b1[ 0 + n0]) * s;
            hr[16 + n0] = (acc1[v] + b1[16 + n0]) * s;
            hr[32 + n0] = (acc2[v] + b1[32 + n0]) * s;
            hr[48 + n0] = (acc3[v] + b1[48 + n0]) * s;
        }
    }
}

// ---- edge scatter: Agg[dst,:] += w_e * Hin[src,:] --------------------------
// One wave per edge, 32 lanes x float2.  src/dst/w forced SGPR-uniform so
// they compile to scalar loads; feature tables (25.6 MB) live in the 192 MB
// L2, so gathers and atomics resolve there.  Edge streams are marked NT to
// avoid evicting the feature tables.
__global__ __launch_bounds__(256) void edge_scatter_kernel(
    const int*   __restrict__ src, const int* __restrict__ dst,
    const float* __restrict__ w,
    const float* __restrict__ Hin,  // [N,64] prescaled
    float*       __restrict__ Agg,  // [N,64] pre-zeroed
    int Ee)
{
    const int lane = threadIdx.x & 31;
    const int wv   = __builtin_amdgcn_readfirstlane(threadIdx.x >> 5);
    const int e    = blockIdx.x * 8 + wv;
    if (e >= Ee) return;                            // wave-uniform
    const int   s  = __builtin_nontemporal_load(src + e);
    const int   d  = __builtin_nontemporal_load(dst + e);
    const float we = __builtin_nontemporal_load(w + e);
    const float2 hv = *(const float2*)(Hin + (size_t)s * H_F + lane * 2);
    float* out = Agg + (size_t)d * H_F + lane * 2;
    atomicAdd(out,     hv.x * we);
    atomicAdd(out + 1, hv.y * we);
}

// ---- fused middle: B *= D_in(g)^{-1/2} * kernel[f] * D_out(ginv)^{-1/2} ----
__global__ __launch_bounds__(256) void gate_norm_kernel(
    float* __restrict__ B,
    const float* __restrict__ rs_din_g,
    const float* __restrict__ rs_dout_i,
    const float* __restrict__ kern,
    int Nn)
{
    int t = blockIdx.x * 256 + threadIdx.x;
    if (t >= Nn * H_F) return;
    const int i = t >> 6;
    const int f = t & 63;
    B[t] *= rs_din_g[i] * rs_dout_i[i] * kern[f];
}

// ---- output head: out = relu(A * D_in(ginv)^{-1/2}) @ W3 + b3 --------------
// One wave per node; lane holds 2 features; wave32 xor-shuffle reduction.
__global__ __launch_bounds__(256) void head_kernel(
    const float* __restrict__ A,
    const float* __restrict__ rs_din_i,
    const float* __restrict__ W3,   // [64,2]
    const float* __restrict__ b3,   // [2]
    float* __restrict__ out,        // [N,2]
    int Nn)
{
    const int lane = threadIdx.x & 31;
    const int wv   = __builtin_amdgcn_readfirstlane(threadIdx.x >> 5);
    const int node = blockIdx.x * 8 + wv;
    if (node >= Nn) return;
    const float2 v = *(const float2*)(A + (size_t)node * H_F + lane * 2);
    const float s  = rs_din_i[node];     // > 0, commutes with relu
    const float x0 = fmaxf(v.x, 0.0f) * s;
    const float x1 = fmaxf(v.y, 0.0f) * s;
    const int f0 = lane * 2;
    float p0 = x0 * W3[f0 * 2 + 0] + x1 * W3[(f0 + 1) * 2 + 0];
    float p1 = x0 * W3[f0 * 2 + 1] + x1 * W3[(f0 + 1) * 2 + 1];
#pragma unroll
    for (int off = 16; off > 0; off >>= 1) {
        p0 += __shfl_xor(p0, off, 32);
        p1 += __shfl_xor(p1, off, 32);
    }
    if (lane == 0) {
        out[node * 2 + 0] = p0 + b3[0];
        out[node * 2 + 1] = p1 + b3[1];
    }
}

// ---------------------------------------------------------------------------
extern "C" void kernel_launch(void* const* d_in, const int* in_sizes, int n_in,
                              void* d_out, int out_size, void* d_ws, size_t ws_size,
                              hipStream_t stream)
{
    const float* in_feat = (const float*)d_in[0];
    const float* W1      = (const float*)d_in[1];
    const float* b1      = (const float*)d_in[2];
    const float* kern    = (const float*)d_in[3];
    const float* W3      = (const float*)d_in[4];
    const float* b3      = (const float*)d_in[5];
    const int*   g_src   = (const int*)d_in[6];
    const int*   g_dst   = (const int*)d_in[7];
    const float* g_w     = (const float*)d_in[8];
    const int*   i_src   = (const int*)d_in[9];
    const int*   i_dst   = (const int*)d_in[10];
    const float* i_w     = (const float*)d_in[11];

    const int Nn = in_sizes[0] / IN_F;   // 100000
    const int Ee = in_sizes[6];          // 1600000

    // workspace layout: [ deg 4N | bufA N*64 | bufB N*64 ]  (~53 MB)
    float* deg  = (float*)d_ws;
    float* bufA = deg + (size_t)4 * Nn;
    float* bufB = bufA + (size_t)Nn * H_F;

    const float* rs_dout_g = deg;
    const float* rs_din_g  = deg + Nn;
    const float* rs_dout_i = deg + 2 * (size_t)Nn;
    const float* rs_din_i  = deg + 3 * (size_t)Nn;

    // 1) degrees -> rsqrt
    hipMemsetAsync(deg, 0, (size_t)4 * Nn * sizeof(float), stream);
    deg_kernel<<<(Ee + 255) / 256, 256, 0, stream>>>(g_src, g_dst, i_src, i_dst, deg, Nn, Ee);
    deg_finish_kernel<<<(4 * Nn + 255) / 256, 256, 0, stream>>>(deg, 4 * Nn);

    // 2) zero agg for gconv1
    hipMemsetAsync(bufB, 0, (size_t)Nn * H_F * sizeof(float), stream);

    // 3) WMMA GEMM with fused bias + D_out(g)^{-1/2}
    const int ntile = (Nn + 15) / 16;
    gemm1_wmma_kernel<<<(ntile + 7) / 8, 256, 0, stream>>>(in_feat, W1, b1, rs_dout_g, bufA, Nn);

    // 4) gconv1 scatter (g): bufB += w_e * bufA[src]
    edge_scatter_kernel<<<(Ee + 7) / 8, 256, 0, stream>>>(g_src, g_dst, g_w, bufA, bufB, Ee);

    // 5) fused: D_in(g)^{-1/2} * kernel * D_out(ginv)^{-1/2}, in place
    gate_norm_kernel<<<(Nn * H_F + 255) / 256, 256, 0, stream>>>(bufB, rs_din_g, rs_dout_i, kern, Nn);

    // 6) zero agg for gconv2 (reuse bufA)
    hipMemsetAsync(bufA, 0, (size_t)Nn * H_F * sizeof(float), stream);

    // 7) gconv2 scatter (ginv): bufA += w_e * bufB[src]
    edge_scatter_kernel<<<(Ee + 7) / 8, 256, 0, stream>>>(i_src, i_dst, i_w, bufB, bufA, Ee);

    // 8) relu + D_in(ginv)^{-1/2} + 64->2 projection
    head_kernel<<<(Nn + 7) / 8, 256, 0, stream>>>(bufA, rs_din_i, W3, b3, (float*)d_out, Nn);
}